// DSALite_88218628260696
// MI455X (gfx1250) — compile-verified
//
#include <hip/hip_runtime.h>

typedef float v2f __attribute__((ext_vector_type(2)));
typedef float v8f __attribute__((ext_vector_type(8)));
typedef int   v4i __attribute__((ext_vector_type(4)));

#define T_DIM 2048
#define D_DIM 128
#define H_DIM 16
#define DS_DIM 256            // T/8
#define KSEL 26               // top-204 over 8x-repeated values -> 26th distinct
#define SCALE_F 0.08838834764831845f   // 1/sqrt(128)
#define NEG_BIG (-1000000000.0f)

#define KS 132                // K/V LDS row stride (floats), 16B-aligned rows
#define PS 18                 // P transpose stride: even (b64 align), conflict-free

// ---------------------------------------------------------------------------
// CDNA5 async global->LDS copy (ASYNCcnt-tracked), with sync fallback.
// Builtin signature (probe-confirmed): params are int4* in AS(1) / AS(3).
// ---------------------------------------------------------------------------
#if defined(__has_builtin)
#if __has_builtin(__builtin_amdgcn_global_load_async_to_lds_b128)
#define HAVE_ASYNC_LDS 1
#endif
#endif

__device__ __forceinline__ void cp16_async(float* lds_dst, const float* gsrc) {
#ifdef HAVE_ASYNC_LDS
  __builtin_amdgcn_global_load_async_to_lds_b128(
      (__attribute__((address_space(1))) v4i*)(gsrc),
      (__attribute__((address_space(3))) v4i*)(lds_dst), 0, 0);
#else
  *(float4*)lds_dst = *(const float4*)gsrc;
#endif
}

__device__ __forceinline__ void wait_async_lds() {
#ifdef HAVE_ASYNC_LDS
#if __has_builtin(__builtin_amdgcn_s_wait_asynccnt)
  __builtin_amdgcn_s_wait_asynccnt(0);
#else
  asm volatile("s_wait_asynccnt 0x0" ::: "memory");
#endif
#endif
}

// ---------------------------------------------------------------------------
// Kernel 1: s_ds[h, i, j] = scale * dot(Q[h, 8i, :], K[h, 8j, :])
// One wave per 16x16 tile; 32 x V_WMMA_F32_16X16X4_F32 over K=128.
// ---------------------------------------------------------------------------
__global__ __launch_bounds__(32)
void dsa_sds_kernel(const float* __restrict__ Qg, const float* __restrict__ Kg,
                    float* __restrict__ sds) {
  const int h  = blockIdx.y;
  const int ti = blockIdx.x >> 4;
  const int tj = blockIdx.x & 15;
  const int l  = threadIdx.x;
  const int lo = l & 15;
  const int hi = l >> 4;

  const int qrow = (ti * 16 + lo) * 8;   // downsampled -> full row
  const int krow = (tj * 16 + lo) * 8;
  const float* qp = Qg + ((size_t)h * T_DIM + qrow) * D_DIM + 2 * hi;
  const float* kp = Kg + ((size_t)h * T_DIM + krow) * D_DIM + 2 * hi;

  v8f c = {};
#pragma unroll
  for (int kc = 0; kc < 8; ++kc) {
    v2f a[4], b[4];
#pragma unroll
    for (int s = 0; s < 4; ++s) {
      a[s] = *(const v2f*)(qp + kc * 16 + s * 4);
      b[s] = *(const v2f*)(kp + kc * 16 + s * 4);
    }
#pragma unroll
    for (int s = 0; s < 4; ++s) {
      c = __builtin_amdgcn_wmma_f32_16x16x4_f32(false, a[s], false, b[s],
                                                (short)0, c, false, false);
    }
  }

  float* op = sds + ((size_t)h * DS_DIM + ti * 16 + 8 * hi) * DS_DIM + tj * 16 + lo;
#pragma unroll
  for (int r = 0; r < 8; ++r)
    op[r * DS_DIM] = c[r] * SCALE_F;
}

// ---------------------------------------------------------------------------
// Kernel 2: t26[row] = 26th largest of sds[row, 0:256].  One wave per row,
// 26 rounds of (local max over 8) + wave-shuffle max + ballot leader removal.
// ---------------------------------------------------------------------------
__global__ __launch_bounds__(256)
void dsa_thr_kernel(const float* __restrict__ sds, float* __restrict__ t26) {
  const int lane = threadIdx.x & 31;
  const int wave = threadIdx.x >> 5;
  const int row  = blockIdx.x * 8 + wave;          // 0 .. H*DS-1
  const float* rp = sds + (size_t)row * DS_DIM;

  float v[8];
#pragma unroll
  for (int j = 0; j < 8; ++j) v[j] = rp[j * 32 + lane];

  float t = NEG_BIG;
  for (int it = 0; it < KSEL; ++it) {
    float lm = v[0]; int li = 0;
#pragma unroll
    for (int j = 1; j < 8; ++j) { if (v[j] > lm) { lm = v[j]; li = j; } }
    float wm = lm;
    wm = fmaxf(wm, __shfl_xor(wm, 1));
    wm = fmaxf(wm, __shfl_xor(wm, 2));
    wm = fmaxf(wm, __shfl_xor(wm, 4));
    wm = fmaxf(wm, __shfl_xor(wm, 8));
    wm = fmaxf(wm, __shfl_xor(wm, 16));
    unsigned long long bal = __ballot(lm == wm);
    int leader = __ffsll(bal) - 1;
    if (lane == leader) v[li] = -3.0e38f;          // remove one instance
    t = wm;
  }
  if (lane == 0) t26[row] = t;
}

// ---------------------------------------------------------------------------
// Kernel 3: flash attention over ALL 2048 columns with exact fp32 bias:
//   j>i : bias = -1e9 ; j<=i : bias = (1 - sigmoid(10*(1+clamp(u_i,0,1))*(sds - t26)))*(-1e9)
// WG = 8 waves x 16 query rows.  K/V tiles double-buffered in LDS via
// GLOBAL_LOAD_ASYNC_TO_LDS_B128 (ASYNCcnt); P transposed via per-wave LDS.
// ---------------------------------------------------------------------------
__global__ __launch_bounds__(256)
void dsa_attn_kernel(const float* __restrict__ Qg, const float* __restrict__ Kg,
                     const float* __restrict__ Vg, const float* __restrict__ Ug,
                     const float* __restrict__ sds, const float* __restrict__ t26g,
                     float* __restrict__ Og) {
  __shared__ float KsL[2][16 * KS];
  __shared__ float VsL[2][16 * KS];
  __shared__ float SsL[16 * DS_DIM];     // sds slice: 16 drows x 256
  __shared__ float PsL[8][16 * PS];      // per-wave P transpose scratch

  const int h   = blockIdx.y;
  const int bx  = blockIdx.x;            // 128-row query stripe
  const int tid = threadIdx.x;
  const int w   = tid >> 5;
  const int l   = tid & 31;
  const int lo  = l & 15;
  const int hi  = l >> 4;

  const float* kg0 = Kg + (size_t)h * T_DIM * D_DIM;
  const float* vg0 = Vg + (size_t)h * T_DIM * D_DIM;

  // per-thread staging coordinates (2 x b128 per tile for K and for V)
  const int srow0 = tid >> 5,         sc0 = (tid & 31) * 4;         // idx = tid
  const int srow1 = (tid + 256) >> 5, sc1 = (tid & 31) * 4;         // idx = tid+256

  // prefetch tile 0 into buffer 0 (async, ASYNCcnt-tracked)
  {
    cp16_async(&KsL[0][srow0 * KS + sc0], kg0 + srow0 * D_DIM + sc0);
    cp16_async(&KsL[0][srow1 * KS + sc1], kg0 + srow1 * D_DIM + sc1);
    cp16_async(&VsL[0][srow0 * KS + sc0], vg0 + srow0 * D_DIM + sc0);
    cp16_async(&VsL[0][srow1 * KS + sc1], vg0 + srow1 * D_DIM + sc1);
  }

  // stage the sds slice for this stripe's 16 downsampled rows
  {
    const float* sp = sds + ((size_t)h * DS_DIM + bx * 16) * DS_DIM;
#pragma unroll
    for (int q = 0; q < 16; ++q) SsL[tid + 256 * q] = sp[tid + 256 * q];
  }

  const int qrow0 = bx * 128 + w * 16;

  // preload Q A-fragments: qa[s] covers k = 4s + 2*hi + {0,1}, row = qrow0+lo
  v2f qa[32];
  {
    const float* qp = Qg + ((size_t)h * T_DIM + qrow0 + lo) * D_DIM + 2 * hi;
#pragma unroll
    for (int s = 0; s < 32; ++s) qa[s] = *(const v2f*)(qp + s * 4);
  }

  const int drl = w * 2 + hi;            // local ds-row for this lane's 8 rows
  const float t26v = t26g[h * DS_DIM + bx * 16 + drl];
  float uu[8];
#pragma unroll
  for (int r = 0; r < 8; ++r) {
    float u = Ug[qrow0 + r + 8 * hi];
    u = fminf(fmaxf(u, 0.0f), 1.0f);
    uu[r] = 1.0f + u;
  }

  v8f acc[8];
  float Mx[8], Ls[8];
  v8f zero = {};
#pragma unroll
  for (int t = 0; t < 8; ++t) acc[t] = zero;
#pragma unroll
  for (int r = 0; r < 8; ++r) { Mx[r] = -3.0e38f; Ls[r] = 0.0f; }

  wait_async_lds();      // tile-0 async copies from this wave complete
  __syncthreads();       // all waves' copies + SsL complete

  const int NT = T_DIM / 16;
  for (int jt = 0; jt < NT; ++jt) {
    const int cur = jt & 1;
    const float* Ksc = KsL[cur];
    const float* Vsc = VsL[cur];

    // async prefetch of next K/V tile into the other buffer
    if (jt + 1 < NT) {
      const float* kgp = kg0 + (size_t)(jt + 1) * 16 * D_DIM;
      const float* vgp = vg0 + (size_t)(jt + 1) * 16 * D_DIM;
      float* kn = (float*)KsL[cur ^ 1];
      float* vn = (float*)VsL[cur ^ 1];
      cp16_async(&kn[srow0 * KS + sc0], kgp + srow0 * D_DIM + sc0);
      cp16_async(&kn[srow1 * KS + sc1], kgp + srow1 * D_DIM + sc1);
      cp16_async(&vn[srow0 * KS + sc0], vgp + srow0 * D_DIM + sc0);
      cp16_async(&vn[srow1 * KS + sc1], vgp + srow1 * D_DIM + sc1);
    }

    // S = Q * K^T  (raw dot, 32 WMMAs over K=128)
    v8f s = zero;
#pragma unroll
    for (int kc = 0; kc < 8; ++kc) {
#pragma unroll
      for (int ss = 0; ss < 4; ++ss) {
        v2f b = *(const v2f*)&Ksc[lo * KS + kc * 16 + ss * 4 + 2 * hi];
        s = __builtin_amdgcn_wmma_f32_16x16x4_f32(false, qa[kc * 4 + ss], false, b,
                                                  (short)0, s, false, false);
      }
    }

    const float sdsv = SsL[drl * DS_DIM + jt * 2 + (lo >> 3)];
    const int jcol = jt * 16 + lo;

    float p[8], corr[8];
#pragma unroll
    for (int r = 0; r < 8; ++r) {
      const int irow = qrow0 + r + 8 * hi;
      float a10  = 10.0f * uu[r] * (sdsv - t26v);
      float omm  = 1.0f - 1.0f / (1.0f + __expf(-a10));   // 1 - sigmoid
      float bias = (jcol <= irow) ? omm * NEG_BIG : NEG_BIG;
      float sc   = s[r] * SCALE_F + bias;
      // row max within the 16-lane half-group
      float m = sc;
      m = fmaxf(m, __shfl_xor(m, 1));
      m = fmaxf(m, __shfl_xor(m, 2));
      m = fmaxf(m, __shfl_xor(m, 4));
      m = fmaxf(m, __shfl_xor(m, 8));
      float nM = fmaxf(Mx[r], m);
      float cf = __expf(Mx[r] - nM);
      float pe = __expf(sc - nM);
      float rs = pe;
      rs += __shfl_xor(rs, 1);
      rs += __shfl_xor(rs, 2);
      rs += __shfl_xor(rs, 4);
      rs += __shfl_xor(rs, 8);
      Ls[r] = Ls[r] * cf + rs;
      Mx[r] = nM;
      corr[r] = cf;
      p[r] = pe;
    }

    // transpose P (C-layout -> A-layout) through per-wave LDS scratch
#pragma unroll
    for (int r = 0; r < 8; ++r)
      PsL[w][(r + 8 * hi) * PS + lo] = p[r];
    asm volatile("s_wait_dscnt 0x0" ::: "memory");

    v2f pa[4];
#pragma unroll
    for (int ss = 0; ss < 4; ++ss)
      pa[ss] = *(const v2f*)&PsL[w][lo * PS + ss * 4 + 2 * hi];

    // O = O*corr + P*V  (8 output 16x16 tiles, 4 WMMAs each over K=16)
#pragma unroll
    for (int t = 0; t < 8; ++t) {
      v8f a2 = acc[t];
#pragma unroll
      for (int r = 0; r < 8; ++r) a2[r] *= corr[r];
#pragma unroll
      for (int ss = 0; ss < 4; ++ss) {
        v2f b;
        b.x = Vsc[(ss * 4 + 2 * hi    ) * KS + t * 16 + lo];
        b.y = Vsc[(ss * 4 + 2 * hi + 1) * KS + t * 16 + lo];
        a2 = __builtin_amdgcn_wmma_f32_16x16x4_f32(false, pa[ss], false, b,
                                                   (short)0, a2, false, false);
      }
      acc[t] = a2;
    }

    wait_async_lds();    // this wave's prefetch landed
    __syncthreads();     // all waves done reading cur + all prefetches visible
  }

  // epilogue: out = acc / L
#pragma unroll
  for (int t = 0; t < 8; ++t) {
    float* op = Og + ((size_t)h * T_DIM + qrow0 + 8 * hi) * D_DIM + t * 16 + lo;
#pragma unroll
    for (int r = 0; r < 8; ++r)
      op[r * D_DIM] = acc[t][r] / Ls[r];
  }
}

// ---------------------------------------------------------------------------
extern "C" void kernel_launch(void* const* d_in, const int* in_sizes, int n_in,
                              void* d_out, int out_size, void* d_ws, size_t ws_size,
                              hipStream_t stream) {
  (void)in_sizes; (void)n_in; (void)out_size; (void)ws_size;
  const float* Q = (const float*)d_in[0];
  const float* K = (const float*)d_in[1];
  const float* V = (const float*)d_in[2];
  const float* U = (const float*)d_in[3];
  float* out = (float*)d_out;

  float* sds = (float*)d_ws;                                   // 16*256*256 floats
  float* t26 = sds + (size_t)H_DIM * DS_DIM * DS_DIM;          // 16*256 floats

  // 1) downsampled scores (scaled)
  dsa_sds_kernel<<<dim3(256, H_DIM), 32, 0, stream>>>(Q, K, sds);
  // 2) per-(head, ds-row) 26th-largest threshold
  dsa_thr_kernel<<<dim3((H_DIM * DS_DIM) / 8), 256, 0, stream>>>(sds, t26);
  // 3) biased flash attention over the full row
  dsa_attn_kernel<<<dim3(T_DIM / 128, H_DIM), 256, 0, stream>>>(Q, K, V, U, sds, t26, out);
}